// GCN_33234456937224
// MI455X (gfx1250) — compile-verified
//
#include <hip/hip_runtime.h>

typedef float v2f __attribute__((ext_vector_type(2)));
typedef float v8f __attribute__((ext_vector_type(8)));

// ---------------------------------------------------------------------------
// Zero a float buffer.
// ---------------------------------------------------------------------------
__global__ void sage_zero_kernel(float* __restrict__ p, int count) {
  int i = blockIdx.x * blockDim.x + threadIdx.x;
  if (i < count) p[i] = 0.0f;
}

// ---------------------------------------------------------------------------
// Layer-1 scatter: d_in = 1.  Also builds the per-destination degree counts
// (reused by every layer).  sums/cnt live in L2 (20 MB total << 192 MB), so
// the atomics never touch HBM.
// ---------------------------------------------------------------------------
__global__ void sage_scatter_d1_kernel(const long long* __restrict__ src,
                                       const long long* __restrict__ dst,
                                       const float* __restrict__ x,
                                       float* __restrict__ sums,
                                       float* __restrict__ cnt,
                                       int E) {
  int e = blockIdx.x * blockDim.x + threadIdx.x;
  if (e >= E) return;
  int s = (int)src[e];
  int d = (int)dst[e];
  atomicAdd(&sums[d], x[s]);
  atomicAdd(&cnt[d], 1.0f);
}

// ---------------------------------------------------------------------------
// d_in = 4 scatter (layers 2 and 3): float4 gather + 4 float atomics.
// ---------------------------------------------------------------------------
__global__ void sage_scatter_d4_kernel(const long long* __restrict__ src,
                                       const long long* __restrict__ dst,
                                       const float* __restrict__ h,
                                       float* __restrict__ sums,
                                       int E) {
  int e = blockIdx.x * blockDim.x + threadIdx.x;
  if (e >= E) return;
  int s = (int)src[e];
  int d = (int)dst[e];
  const float4 v = *(const float4*)(h + 4 * (size_t)s);
  atomicAdd(&sums[4 * d + 0], v.x);
  atomicAdd(&sums[4 * d + 1], v.y);
  atomicAdd(&sums[4 * d + 2], v.z);
  atomicAdd(&sums[4 * d + 3], v.w);
}

// ---------------------------------------------------------------------------
// Layer 1 node transform: din=1 -> dout=4 (scalar; a WMMA with K-dim 1 would
// be pure waste).  h1 = relu(agg*W1l + b1 + x*W1r).
// ---------------------------------------------------------------------------
__global__ void sage_node1_kernel(const float* __restrict__ sums,
                                  const float* __restrict__ cnt,
                                  const float* __restrict__ x,
                                  const float* __restrict__ Wl,
                                  const float* __restrict__ b,
                                  const float* __restrict__ Wr,
                                  float* __restrict__ hout, int N) {
  int i = blockIdx.x * blockDim.x + threadIdx.x;
  if (i >= N) return;
  float inv = 1.0f / fmaxf(cnt[i], 1.0f);
  float agg = sums[i] * inv;
  float xv  = x[i];
  float4 o;
  o.x = fmaxf(agg * Wl[0] + b[0] + xv * Wr[0], 0.0f);
  o.y = fmaxf(agg * Wl[1] + b[1] + xv * Wr[1], 0.0f);
  o.z = fmaxf(agg * Wl[2] + b[2] + xv * Wr[2], 0.0f);
  o.w = fmaxf(agg * Wl[3] + b[3] + xv * Wr[3], 0.0f);
  *(float4*)(hout + 4 * (size_t)i) = o;
}

// ---------------------------------------------------------------------------
// WMMA tile helper for din=4 SAGE transform.
//   D(16x16) = A_agg(16x4) x B_Wl(4x16) + [ A_h(16x4) x B_Wr(4x16) + bias ]
// A f32 16x4 layout: lanes 0-15 hold row M=lane, K={0,1}; lanes 16-31 hold
// row M=lane-16, K={2,3}.  B mirrors it (K split across lane halves, N=lane&15).
// C/D: lane holds column N=lane&15, rows M = g (+8 for upper lanes).
// EXEC is all-ones at both WMMA call sites (tile count divides N).
// ---------------------------------------------------------------------------
template <int DOUT>
__device__ inline v8f sage_wmma_tile(const float* __restrict__ sums,
                                     const float* __restrict__ cnt,
                                     const float* __restrict__ hin,
                                     const float* __restrict__ Wl,
                                     const float* __restrict__ bias,
                                     const float* __restrict__ Wr,
                                     int node_base, int lane) {
  const int n    = lane & 15;          // D column / B column for this lane
  const int kb   = (lane >> 4) << 1;   // K base: 0 or 2
  const int node = node_base + n;      // A row for this lane

  const float inv = 1.0f / fmaxf(cnt[node], 1.0f);
  v2f a_agg, a_h;
  a_agg.x = sums[4 * (size_t)node + kb + 0] * inv;
  a_agg.y = sums[4 * (size_t)node + kb + 1] * inv;
  a_h.x   = hin[4 * (size_t)node + kb + 0];
  a_h.y   = hin[4 * (size_t)node + kb + 1];

  v2f b_l = {0.0f, 0.0f}, b_r = {0.0f, 0.0f};
  float bv = 0.0f;
  if (n < DOUT) {                       // W is [4, DOUT] row-major
    b_l.x = Wl[(kb + 0) * DOUT + n];
    b_l.y = Wl[(kb + 1) * DOUT + n];
    b_r.x = Wr[(kb + 0) * DOUT + n];
    b_r.y = Wr[(kb + 1) * DOUT + n];
    bv    = bias[n];
  }

  v8f acc;
#pragma unroll
  for (int g = 0; g < 8; ++g) acc[g] = bv;     // C = broadcast bias

  acc = __builtin_amdgcn_wmma_f32_16x16x4_f32(
      false, a_agg, false, b_l, (short)0, acc, false, false);
  acc = __builtin_amdgcn_wmma_f32_16x16x4_f32(
      false, a_h, false, b_r, (short)0, acc, false, false);
  return acc;
}

// ---------------------------------------------------------------------------
// Layer 2: din=4 -> dout=4, relu, in-place update of h (each wave reads its 16
// nodes before writing them; blocks are disjoint).  128 threads = 4 waves,
// 16 nodes/wave.
// ---------------------------------------------------------------------------
__global__ void sage_node2_kernel(const float* __restrict__ sums,
                                  const float* __restrict__ cnt,
                                  const float* __restrict__ hin,
                                  const float* __restrict__ Wl,
                                  const float* __restrict__ b,
                                  const float* __restrict__ Wr,
                                  float* __restrict__ hout, int N) {
  __shared__ float tile[4][16][4];
  const int lane = threadIdx.x & 31;
  const int wave = threadIdx.x >> 5;
  const int node_base = (blockIdx.x * 4 + wave) * 16;
  if (node_base >= N) return;

  v8f acc = sage_wmma_tile<4>(sums, cnt, hin, Wl, b, Wr, node_base, lane);

  const int n    = lane & 15;
  const int mofs = (lane < 16) ? 0 : 8;
  if (n < 4) {
#pragma unroll
    for (int g = 0; g < 8; ++g) tile[wave][mofs + g][n] = fmaxf(acc[g], 0.0f);
  }
  __syncthreads();
  if (lane < 16) {
    const int node = node_base + lane;
    float4 o = make_float4(tile[wave][lane][0], tile[wave][lane][1],
                           tile[wave][lane][2], tile[wave][lane][3]);
    *(float4*)(hout + 4 * (size_t)node) = o;
  }
}

// ---------------------------------------------------------------------------
// Layer 3 (din=4 -> dout=2, relu) fused with the final classifier
// (out = h3 @ Wc + bc).  Writes out -> d_out[0 : 2N), h3 -> d_out[2N : 4N).
// ---------------------------------------------------------------------------
__global__ void sage_node3_kernel(const float* __restrict__ sums,
                                  const float* __restrict__ cnt,
                                  const float* __restrict__ hin,
                                  const float* __restrict__ Wl,
                                  const float* __restrict__ b,
                                  const float* __restrict__ Wr,
                                  const float* __restrict__ Wc,
                                  const float* __restrict__ bc,
                                  float* __restrict__ out, int N) {
  __shared__ float tile[4][16][2];
  const int lane = threadIdx.x & 31;
  const int wave = threadIdx.x >> 5;
  const int node_base = (blockIdx.x * 4 + wave) * 16;
  if (node_base >= N) return;

  v8f acc = sage_wmma_tile<2>(sums, cnt, hin, Wl, b, Wr, node_base, lane);

  const int n    = lane & 15;
  const int mofs = (lane < 16) ? 0 : 8;
  if (n < 2) {
#pragma unroll
    for (int g = 0; g < 8; ++g) tile[wave][mofs + g][n] = fmaxf(acc[g], 0.0f);
  }
  __syncthreads();
  if (lane < 16) {
    const int node = node_base + lane;
    const float h0 = tile[wave][lane][0];
    const float h1 = tile[wave][lane][1];
    // out[j] = h0*Wc[0][j] + h1*Wc[1][j] + bc[j]   (Wc is [2,2] row-major)
    out[2 * (size_t)node + 0] = h0 * Wc[0] + h1 * Wc[2] + bc[0];
    out[2 * (size_t)node + 1] = h0 * Wc[1] + h1 * Wc[3] + bc[1];
    out[2 * (size_t)N + 2 * (size_t)node + 0] = h0;
    out[2 * (size_t)N + 2 * (size_t)node + 1] = h1;
  }
}

// ---------------------------------------------------------------------------
// Launcher.  Workspace layout (floats): cnt[N] | sums[4N] | h[4N]  (36 MB).
// ---------------------------------------------------------------------------
extern "C" void kernel_launch(void* const* d_in, const int* in_sizes, int n_in,
                              void* d_out, int out_size, void* d_ws, size_t ws_size,
                              hipStream_t stream) {
  const float*     x   = (const float*)d_in[0];
  const long long* ei  = (const long long*)d_in[1];
  const float*     W1l = (const float*)d_in[2];
  const float*     b1  = (const float*)d_in[3];
  const float*     W1r = (const float*)d_in[4];
  const float*     W2l = (const float*)d_in[5];
  const float*     b2  = (const float*)d_in[6];
  const float*     W2r = (const float*)d_in[7];
  const float*     W3l = (const float*)d_in[8];
  const float*     b3  = (const float*)d_in[9];
  const float*     W3r = (const float*)d_in[10];
  const float*     Wc  = (const float*)d_in[11];
  const float*     bc  = (const float*)d_in[12];

  const int N = in_sizes[0];      // 1,000,000 (divisible by 16)
  const int E = in_sizes[1] / 2;  // 16,000,000
  const long long* src = ei;
  const long long* dst = ei + (size_t)E;

  float* ws   = (float*)d_ws;
  float* cnt  = ws;                    // N
  float* sums = ws + (size_t)N;        // 4N
  float* h    = ws + 5 * (size_t)N;    // 4N, updated in place per layer

  float* out = (float*)d_out;

  const dim3 B256(256), B128(128);
  const int gZero5 = (5 * N + 255) / 256;
  const int gZero4 = (4 * N + 255) / 256;
  const int gEdge  = (E + 255) / 256;
  const int gNode  = (N + 255) / 256;
  const int gTile  = (N + 63) / 64;    // 4 waves/block * 16 nodes/wave

  // Layer 1
  sage_zero_kernel<<<gZero5, B256, 0, stream>>>(ws, 5 * N);   // cnt + sums
  sage_scatter_d1_kernel<<<gEdge, B256, 0, stream>>>(src, dst, x, sums, cnt, E);
  sage_node1_kernel<<<gNode, B256, 0, stream>>>(sums, cnt, x, W1l, b1, W1r, h, N);

  // Layer 2
  sage_zero_kernel<<<gZero4, B256, 0, stream>>>(sums, 4 * N);
  sage_scatter_d4_kernel<<<gEdge, B256, 0, stream>>>(src, dst, h, sums, E);
  sage_node2_kernel<<<gTile, B128, 0, stream>>>(sums, cnt, h, W2l, b2, W2r, h, N);

  // Layer 3 + classifier
  sage_zero_kernel<<<gZero4, B256, 0, stream>>>(sums, 4 * N);
  sage_scatter_d4_kernel<<<gEdge, B256, 0, stream>>>(src, dst, h, sums, E);
  sage_node3_kernel<<<gTile, B128, 0, stream>>>(sums, cnt, h, W3l, b3, W3r, Wc, bc, out, N);
}